// AccFlow2FrameEncoder_16836271800627
// MI455X (gfx1250) — compile-verified
//
#include <hip/hip_runtime.h>
#include <hip/hip_bf16.h>

// ---------------------------------------------------------------------------
// DynamicPillarFeatureNet 2-frame diff for MI455X (gfx1250).
// Memory-bound (~300 MiB @ 23.3 TB/s => ~13us floor). The 9->64 GEMM runs on
// the matrix pipe via V_WMMA_F32_16X16X4_F32 (fp32 required: bf16's 8-bit
// mantissa would quantize +-51.2m coords by ~0.2m). BN folded into W + C-init.
// Scatter is branch-free: invalid tail lanes contribute 0.0f atomics.
// ---------------------------------------------------------------------------

typedef float v2f __attribute__((ext_vector_type(2)));
typedef float v8f __attribute__((ext_vector_type(8)));

#define GXD   512
#define GYD   512
#define NVOX  (GXD * GYD)
#define CCH   64
#define NBATCH 2
#define VXS   0.2f
#define XMINF (-51.2f)
#define EPSBN 1e-3f

__device__ __forceinline__ int voxel_of(float x, float y) {
  int ix = (int)floorf((x - XMINF) / VXS);
  int iy = (int)floorf((y - XMINF) / VXS);
  ix = min(max(ix, 0), GXD - 1);
  iy = min(max(iy, 0), GYD - 1);
  return ix * GYD + iy;
}

// ---------------------------------------------------------------------------
// Pass 1: per-(cloud,batch,voxel) point count and xyz sum (for cluster mean).
// ---------------------------------------------------------------------------
__global__ void pillar_count(const float* __restrict__ pc0,
                             const float* __restrict__ pc1,
                             float* __restrict__ cnt,    // [2][B][NVOX]
                             float* __restrict__ csum,   // [2][B][NVOX][3]
                             int npts) {
  const long total = 2L * NBATCH * npts;
  for (long i = (long)blockIdx.x * blockDim.x + threadIdx.x; i < total;
       i += (long)gridDim.x * blockDim.x) {
    int  cloud = (int)(i / ((long)NBATCH * npts));
    long rem   = i - (long)cloud * NBATCH * npts;
    int  b     = (int)(rem / npts);
    int  p     = (int)(rem - (long)b * npts);
    const float* q = (cloud ? pc1 : pc0) + ((size_t)b * npts + p) * 3;
    __builtin_prefetch(q + 3 * 256, 0, 1);   // global_prefetch_b8
    float x = q[0], y = q[1], z = q[2];
    int vid = voxel_of(x, y);
    size_t seg = ((size_t)cloud * NBATCH + b) * NVOX + vid;
    atomicAdd(&cnt[seg], 1.0f);
    atomicAdd(&csum[seg * 3 + 0], x);
    atomicAdd(&csum[seg * 3 + 1], y);
    atomicAdd(&csum[seg * 3 + 2], z);
  }
}

// ---------------------------------------------------------------------------
// Pass 2: features -> fp32 WMMA GEMM (9->64, K padded to 12) -> ReLU ->
// branch-free atomic scatter-add into per-voxel channel sums.
// Block = 256 threads = 8 waves; each wave owns 32 points (2 M-tiles of 16).
// ---------------------------------------------------------------------------
__global__ void pillar_feat(const float* __restrict__ pc0,
                            const float* __restrict__ pc1,
                            const float* __restrict__ W,      // [9][64]
                            const float* __restrict__ gamma,
                            const float* __restrict__ beta,
                            const float* __restrict__ mean,
                            const float* __restrict__ var,
                            const float* __restrict__ cnt,    // [2][B][NVOX]
                            const float* __restrict__ csum,   // [2][B][NVOX][3]
                            float* __restrict__ fsum0,        // [B][NVOX][64]
                            float* __restrict__ fsum1,        // == d_out
                            int npts) {
  __shared__ float sScale[CCH];
  __shared__ float sBias[CCH];
  __shared__ float sW[12][CCH];          // BN-folded, K padded with zeros
  __shared__ float sF[8][32][12];        // per-wave point features
  __shared__ int   sVid[8][32];
  __shared__ float sValF[8][32];         // 1.0 valid / 0.0 tail-clamped

  const int cloud = blockIdx.z;
  const int b     = blockIdx.y;
  const int tid   = threadIdx.x;
  const int wave  = tid >> 5;
  const int lane  = tid & 31;

  // ---- fold BN into weights -----------------------------------------------
  if (tid < CCH) {
    float sc = gamma[tid] * rsqrtf(var[tid] + EPSBN);
    sScale[tid] = sc;
    sBias[tid]  = beta[tid] - mean[tid] * sc;
  }
  __syncthreads();
  for (int e = tid; e < 12 * CCH; e += blockDim.x) {
    int k = e >> 6, c = e & 63;
    sW[k][c] = (k < 9) ? W[k * CCH + c] * sScale[c] : 0.0f;
  }

  // ---- per-point 9-feature augmentation -----------------------------------
  const int  pbase = blockIdx.x * 256 + wave * 32;
  const int  p     = pbase + lane;
  const bool valid = (p < npts);
  const int  pcl   = valid ? p : (npts - 1);
  const float* q = (cloud ? pc1 : pc0) + ((size_t)b * npts + pcl) * 3;
  float x = q[0], y = q[1], z = q[2];
  int vid = voxel_of(x, y);
  size_t seg = ((size_t)cloud * NBATCH + b) * NVOX + vid;
  float c_n  = fmaxf(cnt[seg], 1.0f);
  float cmx = csum[seg * 3 + 0] / c_n;
  float cmy = csum[seg * 3 + 1] / c_n;
  float cmz = csum[seg * 3 + 2] / c_n;
  float cx = ((float)(vid / GYD) + 0.5f) * VXS + XMINF;
  float cy = ((float)(vid % GYD) + 0.5f) * VXS + XMINF;
  float f[12];
  f[0] = x;        f[1] = y;        f[2] = z;
  f[3] = x - cmx;  f[4] = y - cmy;  f[5] = z - cmz;
  f[6] = x - cx;   f[7] = y - cy;   f[8] = z;        // Z_CENTER == 0
  f[9] = 0.f; f[10] = 0.f; f[11] = 0.f;
#pragma unroll
  for (int k = 0; k < 12; ++k) sF[wave][lane][k] = f[k];
  sVid[wave][lane]  = vid;
  sValF[wave][lane] = valid ? 1.0f : 0.0f;
  __syncthreads();

  // ---- WMMA fp32 16x16x4: D = A(16x4) * B(4x16) + C -----------------------
  // A frag (ISA 7.12.2): lanes 0-15 -> M=0..15; VGPR0:{K=0|K=2}, VGPR1:{K=1|K=3}
  const int m16 = lane & 15;
  const int kh  = lane >> 4;                 // selects K pair (+0 / +2)
  float* dst = ((cloud == 0) ? fsum0 : fsum1) + (size_t)b * NVOX * CCH;

  // B fragments + bias are M-tile invariant: keep them in registers.
  v2f   wf[4][3];
  float biasv[4];
#pragma unroll
  for (int nt = 0; nt < 4; ++nt) {
    const int n = nt * 16 + m16;
    biasv[nt] = sBias[n];
#pragma unroll
    for (int kc = 0; kc < 3; ++kc) {
      const int k0 = kc * 4 + 2 * kh;
      wf[nt][kc].x = sW[k0][n];
      wf[nt][kc].y = sW[k0 + 1][n];
    }
  }

#pragma unroll
  for (int mt = 0; mt < 2; ++mt) {
    // A fragments for this M-tile.
    v2f af[3];
#pragma unroll
    for (int kc = 0; kc < 3; ++kc) {
      const int k0 = kc * 4 + 2 * kh;
      af[kc].x = sF[wave][mt * 16 + m16][k0];
      af[kc].y = sF[wave][mt * 16 + m16][k0 + 1];
    }
    // Row metadata (C layout: lane L, reg r -> row M = r + 8*(L>=16)).
    float* rowp[8];
    float  rowf[8];
#pragma unroll
    for (int r = 0; r < 8; ++r) {
      const int pl = mt * 16 + 8 * kh + r;
      rowp[r] = dst + (size_t)sVid[wave][pl] * CCH;
      rowf[r] = sValF[wave][pl];
    }
#pragma unroll
    for (int nt = 0; nt < 4; ++nt) {
      const int n = nt * 16 + m16;           // this lane's output channel
      v8f acc;
#pragma unroll
      for (int r = 0; r < 8; ++r) acc[r] = biasv[nt];
#pragma unroll
      for (int kc = 0; kc < 3; ++kc) {       // K = 12 in chunks of 4
        acc = __builtin_amdgcn_wmma_f32_16x16x4_f32(
            /*neg_a=*/false, af[kc], /*neg_b=*/false, wf[nt][kc],
            /*c_mod=*/(short)0, acc, /*reuse_a=*/false, /*reuse_b=*/false);
      }
      // ReLU + branch-free scatter (tail lanes add 0.0f).
#pragma unroll
      for (int r = 0; r < 8; ++r) {
        float v = fmaxf(acc[r], 0.0f) * rowf[r];
        atomicAdd(rowp[r] + n, v);
      }
    }
  }
}

// ---------------------------------------------------------------------------
// Pass 3: out = fsum1/max(c1,1) - fsum0/max(c0,1), float4-vectorized.
// ---------------------------------------------------------------------------
__global__ void pillar_finalize(const float* __restrict__ cnt,   // [2][B][NVOX]
                                const float* __restrict__ fsum0, // [B][NVOX][64]
                                float* __restrict__ out) {       // holds fsum1
  const long total4 = (long)NBATCH * NVOX * CCH / 4;
  for (long i = (long)blockIdx.x * blockDim.x + threadIdx.x; i < total4;
       i += (long)gridDim.x * blockDim.x) {
    const long bv = (i * 4) >> 6;                        // b*NVOX + vox
    float r0 = 1.0f / fmaxf(cnt[bv], 1.0f);              // cloud 0
    float r1 = 1.0f / fmaxf(cnt[(size_t)NBATCH * NVOX + bv], 1.0f);
    float4 s1 = ((const float4*)out)[i];
    float4 s0 = ((const float4*)fsum0)[i];
    float4 d;
    d.x = s1.x * r1 - s0.x * r0;
    d.y = s1.y * r1 - s0.y * r0;
    d.z = s1.z * r1 - s0.z * r0;
    d.w = s1.w * r1 - s0.w * r0;
    ((float4*)out)[i] = d;
  }
}

// ---------------------------------------------------------------------------
extern "C" void kernel_launch(void* const* d_in, const int* in_sizes, int n_in,
                              void* d_out, int out_size, void* d_ws, size_t ws_size,
                              hipStream_t stream) {
  const float* pc0   = (const float*)d_in[0];
  const float* pc1   = (const float*)d_in[1];
  const float* W     = (const float*)d_in[2];
  const float* gamma = (const float*)d_in[3];
  const float* beta  = (const float*)d_in[4];
  const float* mean  = (const float*)d_in[5];
  const float* var   = (const float*)d_in[6];
  float*       out   = (float*)d_out;
  float*       ws    = (float*)d_ws;

  const int npts = in_sizes[0] / (NBATCH * 3);   // 100000

  // workspace layout (floats)
  const size_t cntElems  = 2ULL * NBATCH * NVOX;          // counts, both clouds
  float* cnt   = ws;
  float* csum  = cnt + cntElems;                          // xyz sums
  float* fsum0 = csum + cntElems * 3;                     // cloud-0 channel sums
  const size_t zeroFloats = cntElems * 4 + (size_t)NBATCH * NVOX * CCH;

  hipMemsetAsync(d_ws, 0, zeroFloats * sizeof(float), stream);
  hipMemsetAsync(d_out, 0, (size_t)out_size * sizeof(float), stream); // fsum1 acc

  pillar_count<<<1024, 256, 0, stream>>>(pc0, pc1, cnt, csum, npts);

  dim3 fgrid((npts + 255) / 256, NBATCH, 2);
  pillar_feat<<<fgrid, 256, 0, stream>>>(pc0, pc1, W, gamma, beta, mean, var,
                                         cnt, csum, fsum0, out, npts);

  pillar_finalize<<<4096, 256, 0, stream>>>(cnt, fsum0, out);
}